// QloraLinearDebug_14121852469693
// MI455X (gfx1250) — compile-verified
//
#include <hip/hip_runtime.h>

typedef __attribute__((ext_vector_type(2))) float v2f;
typedef __attribute__((ext_vector_type(8))) float v8f;
typedef int v4i_t __attribute__((vector_size(4 * sizeof(int))));

#define IN_F   4096
#define KFOLD  1024
#define OUT_F  4096
#define RANK   16
#define M_TOT  8192

// ---------------------------------------------------------------------------
// Pass 1a: fold x over the 4 tiled copies of the packed weight's K dimension.
// xf[m, k0] = x[m,k0] + x[m,k0+1024] + x[m,k0+2048] + x[m,k0+3072]
// ---------------------------------------------------------------------------
__global__ __launch_bounds__(256) void qlora_fold_kernel(const float* __restrict__ x,
                                                         float* __restrict__ xf) {
  int idx = blockIdx.x * 256 + threadIdx.x;       // one float4 of xf per thread
  int m = idx >> 8;                               // 256 float4 per folded row
  int j = (idx & 255) << 2;                       // k0
  const float* xr = x + (size_t)m * IN_F + j;
  float4 a = *(const float4*)(xr);
  float4 b = *(const float4*)(xr + 1024);
  float4 c = *(const float4*)(xr + 2048);
  float4 d = *(const float4*)(xr + 3072);
  float4 r;
  r.x = (a.x + b.x) + (c.x + d.x);
  r.y = (a.y + b.y) + (c.y + d.y);
  r.z = (a.z + b.z) + (c.z + d.z);
  r.w = (a.w + b.w) + (c.w + d.w);
  *(float4*)(xf + (size_t)m * KFOLD + j) = r;
}

// ---------------------------------------------------------------------------
// Pass 1b: t[m, r] = scaling * sum_k x[m,k] * lora_A[r,k]   (rank-16, K=4096)
// One wave handles 4 rows; lanes stride K; A values reused across the 4 rows.
// Cross-lane xor-tree reduction, then lanes 0..15 store t[m, 0..15].
// Scaling is pre-applied here so the GEMM epilogue can DMA lora_B untouched.
// ---------------------------------------------------------------------------
__global__ __launch_bounds__(256) void qlora_lora_t_kernel(const float* __restrict__ x,
                                                           const float* __restrict__ A,
                                                           const int* __restrict__ scalep,
                                                           float* __restrict__ t) {
  int tid  = threadIdx.x;
  int wave = tid >> 5;
  int lane = tid & 31;
  int m0   = blockIdx.x * 32 + wave * 4;          // 4 rows per wave

  float tp[4][RANK];
#pragma unroll
  for (int row = 0; row < 4; ++row)
#pragma unroll
    for (int r = 0; r < RANK; ++r) tp[row][r] = 0.f;

  const float* x0 = x + (size_t)(m0 + 0) * IN_F;
  const float* x1 = x + (size_t)(m0 + 1) * IN_F;
  const float* x2 = x + (size_t)(m0 + 2) * IN_F;
  const float* x3 = x + (size_t)(m0 + 3) * IN_F;

  for (int k = lane; k < IN_F; k += 32) {
    float v0 = x0[k], v1 = x1[k], v2 = x2[k], v3 = x3[k];
#pragma unroll
    for (int r = 0; r < RANK; ++r) {
      float av = A[r * IN_F + k];
      tp[0][r] = fmaf(v0, av, tp[0][r]);
      tp[1][r] = fmaf(v1, av, tp[1][r]);
      tp[2][r] = fmaf(v2, av, tp[2][r]);
      tp[3][r] = fmaf(v3, av, tp[3][r]);
    }
  }

#pragma unroll
  for (int row = 0; row < 4; ++row)
#pragma unroll
    for (int r = 0; r < RANK; ++r) {
      float v = tp[row][r];
      v += __shfl_xor(v, 16, 32);
      v += __shfl_xor(v, 8, 32);
      v += __shfl_xor(v, 4, 32);
      v += __shfl_xor(v, 2, 32);
      v += __shfl_xor(v, 1, 32);
      tp[row][r] = v;   // every lane now holds the full sum
    }

  float s = (float)(*scalep);
#pragma unroll
  for (int row = 0; row < 4; ++row) {
    float sel = tp[row][0];
#pragma unroll
    for (int r = 1; r < RANK; ++r) sel = (lane == r) ? tp[row][r] : sel;
    if (lane < RANK) t[(size_t)(m0 + row) * RANK + lane] = sel * s;
  }
}

// ---------------------------------------------------------------------------
// Pass 2: out = xf @ Wp^T  (M=8192, N=4096, K=1024)  +  t @ B^T (K=16)
// Workgroup tile 128(M) x 256(N); 8 waves in a 2x4 grid, each wave owns a
// 64x64 block = 4x4 WMMA f32 16x16x4 subtiles. K double-buffered through LDS
// in 16-wide chunks via GLOBAL_LOAD_ASYNC_TO_LDS_B128 (ASYNCcnt path).
// Pitch 20 floats: 16 distinct banks for 16-row frag reads, 8B-aligned rows.
// ---------------------------------------------------------------------------
#define BM 128
#define BN 256
#define BK 16
#define LDP 20

__device__ __forceinline__ void async_copy16(const float* g, float* l) {
#if defined(__gfx1250__) && __has_builtin(__builtin_amdgcn_global_load_async_to_lds_b128)
  __builtin_amdgcn_global_load_async_to_lds_b128(
      (__attribute__((address_space(1))) v4i_t*)g,
      (__attribute__((address_space(3))) v4i_t*)l,
      0, 0);
#else
  *(float4*)l = *(const float4*)g;
#endif
}

__device__ __forceinline__ void wait_async0() {
#if defined(__gfx1250__) && __has_builtin(__builtin_amdgcn_s_wait_asynccnt)
  __builtin_amdgcn_s_wait_asynccnt(0);
#elif defined(__gfx1250__)
  asm volatile("s_wait_asynccnt 0x0" ::: "memory");
#endif
}

__global__ __launch_bounds__(256) void qlora_gemm_kernel(const float* __restrict__ xf,
                                                         const float* __restrict__ Wp,
                                                         const float* __restrict__ tmat,
                                                         const float* __restrict__ loraB,
                                                         float* __restrict__ out) {
  __shared__ float Abuf[2][BM * LDP];   // 2 x 10 KiB
  __shared__ float Bbuf[2][BN * LDP];   // 2 x 20 KiB   (total 60 KiB)

  int tid  = threadIdx.x;
  int wave = tid >> 5;
  int lane = tid & 31;
  int lane15 = lane & 15;
  int khalf  = (lane >> 4) << 1;          // 0 for lanes 0-15, 2 for 16-31

  int n0 = blockIdx.x * BN;
  int m0 = blockIdx.y * BM;
  int wm = wave >> 2;                     // 0..1  (M direction, 64 rows each)
  int wn = wave & 3;                      // 0..3  (N direction, 64 cols each)

  // staging coordinates: 16-float rows, one float4 per slot
  int sRow = tid >> 2;                    // 0..63
  int sCol = (tid & 3) << 2;              // 0,4,8,12

  v8f acc[4][4];
#pragma unroll
  for (int im = 0; im < 4; ++im)
#pragma unroll
    for (int in = 0; in < 4; ++in) acc[im][in] = (v8f)0.f;

  int aOff[4], bOff[4];
#pragma unroll
  for (int i = 0; i < 4; ++i) {
    aOff[i] = (wm * 64 + i * 16 + lane15) * LDP;
    bOff[i] = (wn * 64 + i * 16 + lane15) * LDP;
  }

  // ---- stage chunk 0 into buffer 0 ----
#pragma unroll
  for (int i = 0; i < 2; ++i) {           // A: 128 rows x 16
    int row = sRow + i * 64;
    async_copy16(xf + (size_t)(m0 + row) * KFOLD + sCol, &Abuf[0][row * LDP + sCol]);
  }
#pragma unroll
  for (int i = 0; i < 4; ++i) {           // B: 256 rows x 16
    int row = sRow + i * 64;
    async_copy16(Wp + (size_t)(n0 + row) * KFOLD + sCol, &Bbuf[0][row * LDP + sCol]);
  }

  const int NCHUNK = KFOLD / BK;          // 64
  for (int c = 0; c < NCHUNK; ++c) {
    int buf = c & 1;
    wait_async0();
    __syncthreads();                      // chunk c resident; all waves done with buf^1

    if (c + 1 < NCHUNK) {
      int kb = (c + 1) * BK;
#pragma unroll
      for (int i = 0; i < 2; ++i) {
        int row = sRow + i * 64;
        async_copy16(xf + (size_t)(m0 + row) * KFOLD + kb + sCol,
                     &Abuf[buf ^ 1][row * LDP + sCol]);
      }
#pragma unroll
      for (int i = 0; i < 4; ++i) {
        int row = sRow + i * 64;
        async_copy16(Wp + (size_t)(n0 + row) * KFOLD + kb + sCol,
                     &Bbuf[buf ^ 1][row * LDP + sCol]);
      }
    }

    const float* As = Abuf[buf];
    const float* Bs = Bbuf[buf];
#pragma unroll
    for (int k4 = 0; k4 < BK; k4 += 4) {
      int col = k4 + khalf;
      v2f a[4], b[4];
#pragma unroll
      for (int im = 0; im < 4; ++im) a[im] = *(const v2f*)(&As[aOff[im] + col]);
#pragma unroll
      for (int in = 0; in < 4; ++in) b[in] = *(const v2f*)(&Bs[bOff[in] + col]);
#pragma unroll
      for (int im = 0; im < 4; ++im)
#pragma unroll
        for (int in = 0; in < 4; ++in)
          acc[im][in] = __builtin_amdgcn_wmma_f32_16x16x4_f32(
              false, a[im], false, b[in], (short)0, acc[im][in], false, false);
    }
  }

  // ---- fused LoRA epilogue: acc += (s*t) @ loraB^T  (K = 16 = BK) ----
  // buffer 0 was last read at chunk 62; every wave passed the chunk-63 barrier
  // after that, so it is safe to overwrite without another barrier.
#pragma unroll
  for (int i = 0; i < 2; ++i) {           // 128 x 16 t tile (pre-scaled)
    int row = sRow + i * 64;
    async_copy16(tmat + (size_t)(m0 + row) * RANK + sCol, &Abuf[0][row * LDP + sCol]);
  }
#pragma unroll
  for (int i = 0; i < 4; ++i) {           // 256 x 16 loraB tile
    int row = sRow + i * 64;
    async_copy16(loraB + (size_t)(n0 + row) * RANK + sCol, &Bbuf[0][row * LDP + sCol]);
  }
  wait_async0();
  __syncthreads();

#pragma unroll
  for (int k4 = 0; k4 < RANK; k4 += 4) {
    int col = k4 + khalf;
    v2f a[4], b[4];
#pragma unroll
    for (int im = 0; im < 4; ++im) a[im] = *(const v2f*)(&Abuf[0][aOff[im] + col]);
#pragma unroll
    for (int in = 0; in < 4; ++in) b[in] = *(const v2f*)(&Bbuf[0][bOff[in] + col]);
#pragma unroll
    for (int im = 0; im < 4; ++im)
#pragma unroll
      for (int in = 0; in < 4; ++in)
        acc[im][in] = __builtin_amdgcn_wmma_f32_16x16x4_f32(
            false, a[im], false, b[in], (short)0, acc[im][in], false, false);
  }

  // ---- store: VGPR i holds rows {i, i+8} (by lane half), cols = lane15 ----
  int rhalf = (lane >> 4) << 3;           // 0 or 8
#pragma unroll
  for (int im = 0; im < 4; ++im) {
    int rbase = m0 + wm * 64 + im * 16 + rhalf;
#pragma unroll
    for (int in = 0; in < 4; ++in) {
      int colg = n0 + wn * 64 + in * 16 + lane15;
#pragma unroll
      for (int i = 0; i < 8; ++i)
        out[(size_t)(rbase + i) * OUT_F + colg] = acc[im][in][i];
    }
  }
}

// ---------------------------------------------------------------------------
extern "C" void kernel_launch(void* const* d_in, const int* in_sizes, int n_in,
                              void* d_out, int out_size, void* d_ws, size_t ws_size,
                              hipStream_t stream) {
  const float* x      = (const float*)d_in[0];   // [4,2048,4096]
  const float* Wp     = (const float*)d_in[1];   // [4096,1024]
  const float* loraA  = (const float*)d_in[2];   // [16,4096]
  const float* loraB  = (const float*)d_in[3];   // [4096,16]
  const int*   scalep = (const int*)d_in[4];     // scalar
  float* out = (float*)d_out;

  float* xf = (float*)d_ws;                             // 8192*1024 floats = 32 MiB
  float* t  = xf + (size_t)M_TOT * KFOLD;               // 8192*16 floats  = 512 KiB

  // 1a: fold x (elementwise, bandwidth-bound)
  qlora_fold_kernel<<<(M_TOT * (KFOLD / 4)) / 256, 256, 0, stream>>>(x, xf);
  // 1b: t = scaling * (x @ A^T)
  qlora_lora_t_kernel<<<M_TOT / 32, 256, 0, stream>>>(x, loraA, scalep, t);
  // 2: out = xf @ Wp^T + t @ B^T  (WMMA f32, async double-buffered LDS)
  dim3 grid(OUT_F / BN, M_TOT / BM);
  qlora_gemm_kernel<<<grid, 256, 0, stream>>>(xf, Wp, t, loraB, out);
}